// Mamba_49838800503533
// MI455X (gfx1250) — compile-verified
//
#include <hip/hip_runtime.h>
#include <math.h>

// Problem constants (match reference)
#define B_     2
#define L_     1024
#define DM     768
#define ED     1536
#define NST    64
#define DTR    12
#define DBCW   (DTR + 2*NST)   // 140
#define MROWS  (B_*L_)         // 2048

typedef float v2f __attribute__((ext_vector_type(2)));
typedef float v8f __attribute__((ext_vector_type(8)));

// -----------------------------------------------------------------------------
// fp32 WMMA GEMM: C[M,N] = A[M,K] * B[K,N], row-major, beta=0.
// Block tile 128x128, 256 threads = 8 waves (wave32). K in LDS chunks of 16,
// double-buffered with register staging. Wave grid 2(M) x 4(N); each wave owns
// a 64x32 region = 4x2 tiles of 16x16 via V_WMMA_F32_16X16X4_F32.
//
// LDS tiles stored K-minor ([m][k] / [n][k], row stride 20 floats) so each
// fragment (two consecutive K values per lane) is ONE ds_load_b64 into an
// even-aligned VGPR pair: no packing movs. Stride 20: 8B/16B aligned, and
// 20*lane mod 64 banks are distinct over 16 lanes (gcd(20,64)=4, cycle 16);
// the upper lane half is offset by +2 words -> disjoint bank sets.
//
// Fragment layouts per CDNA5 ISA 7.12.2 (wave32):
//   A 16x4 f32 : lane l holds M = l%16; VGPR j holds K = (l/16)*2 + j
//   B 4x16 f32 : lane l holds N = l%16; VGPR j holds K = (l/16)*2 + j
//   C/D 16x16  : lane l holds N = l%16; VGPR v holds M = v + 8*(l/16)
// -----------------------------------------------------------------------------
#define BM 128
#define BN 128
#define KC 16
#define GT 256   // threads per block

__device__ __forceinline__ void stage_load(const float* __restrict__ Ag,
                                           const float* __restrict__ Bg,
                                           int M, int N, int K,
                                           int bm, int bn, int k0, int tid,
                                           float4* ga, float* gb) {
  // A: 128 rows x 16 k = 512 float4s, 2 per thread, coalesced b128
  #pragma unroll
  for (int i = 0; i < 2; ++i) {
    int idx  = tid + i * GT;
    int mm   = idx >> 2;
    int quad = idx & 3;
    int gm   = bm + mm;
    if (gm < M) ga[i] = *(const float4*)(Ag + (size_t)gm * K + k0 + quad * 4);
    else        ga[i] = (float4){0.f, 0.f, 0.f, 0.f};
  }
  // B: 16 k-rows x 128 n, coalesced b32 in n
  #pragma unroll
  for (int i = 0; i < 8; ++i) {
    int idx = tid + i * GT;
    int nn  = idx & 127;
    int kk  = idx >> 7;
    int gn  = bn + nn;
    gb[i] = (gn < N) ? Bg[(size_t)(k0 + kk) * N + gn] : 0.f;
  }
}

__device__ __forceinline__ void stage_store(float (*As)[20], float (*Bs)[20],
                                            int tid, const float4* ga,
                                            const float* gb) {
  #pragma unroll
  for (int i = 0; i < 2; ++i) {
    int idx  = tid + i * GT;
    int mm   = idx >> 2;
    int quad = idx & 3;
    *(float4*)(&As[mm][quad * 4]) = ga[i];       // 16B-aligned (stride 20)
  }
  #pragma unroll
  for (int i = 0; i < 8; ++i) {
    int idx = tid + i * GT;
    int nn  = idx & 127;
    int kk  = idx >> 7;
    Bs[nn][kk] = gb[i];                          // transpose-on-store
  }
}

__global__ __launch_bounds__(GT)
void gemm_f32_wmma(const float* __restrict__ Ag, const float* __restrict__ Bg,
                   float* __restrict__ Cg, int M, int N, int K) {
  __shared__ float As[2][BM][20];
  __shared__ float Bs[2][BN][20];

  const int tid   = threadIdx.x;
  const int lane  = tid & 31;
  const int wave  = tid >> 5;
  const int wm    = (wave >> 2) * 64;       // 0 or 64
  const int wn    = (wave & 3)  * 32;       // 0,32,64,96
  const int bm    = blockIdx.y * BM;
  const int bn    = blockIdx.x * BN;
  const int mloc  = lane & 15;
  const int kbase = (lane >> 4) * 2;

  v8f acc[4][2];
  #pragma unroll
  for (int i = 0; i < 4; ++i)
    #pragma unroll
    for (int j = 0; j < 2; ++j)
      acc[i][j] = (v8f){0.f,0.f,0.f,0.f,0.f,0.f,0.f,0.f};

  float4 ga[2];
  float  gb[8];
  const int NC = K / KC;

  // Prologue: chunk 0 into buffer 0
  stage_load(Ag, Bg, M, N, K, bm, bn, 0, tid, ga, gb);
  stage_store(As[0], Bs[0], tid, ga, gb);
  __syncthreads();

  for (int c = 0; c < NC; ++c) {
    const int cur = c & 1;
    if (c + 1 < NC)  // issue next chunk's global loads before compute
      stage_load(Ag, Bg, M, N, K, bm, bn, (c + 1) * KC, tid, ga, gb);

    #pragma unroll
    for (int kk = 0; kk < KC; kk += 4) {
      const int kr = kk + kbase;
      v2f av[4], bv[2];
      #pragma unroll
      for (int im = 0; im < 4; ++im)
        av[im] = *(const v2f*)(&As[cur][wm + im * 16 + mloc][kr]);
      #pragma unroll
      for (int in = 0; in < 2; ++in)
        bv[in] = *(const v2f*)(&Bs[cur][wn + in * 16 + mloc][kr]);
      #pragma unroll
      for (int im = 0; im < 4; ++im)
        #pragma unroll
        for (int in = 0; in < 2; ++in)
          acc[im][in] = __builtin_amdgcn_wmma_f32_16x16x4_f32(
              false, av[im], false, bv[in], (short)0, acc[im][in],
              false, false);
    }

    if (c + 1 < NC) {
      __syncthreads();
      stage_store(As[1 - cur], Bs[1 - cur], tid, ga, gb);
      __syncthreads();
    }
  }

  // Store 4x2 tiles of 16x16
  const int nloc   = lane & 15;
  const int rowsel = (lane >> 4) * 8;
  #pragma unroll
  for (int im = 0; im < 4; ++im) {
    #pragma unroll
    for (int in = 0; in < 2; ++in) {
      int gn = bn + wn + in * 16 + nloc;
      if (gn < N) {
        #pragma unroll
        for (int v = 0; v < 8; ++v) {
          int gm = bm + wm + im * 16 + rowsel + v;
          if (gm < M) Cg[(size_t)gm * N + gn] = acc[im][in][v];
        }
      }
    }
  }
}

// -----------------------------------------------------------------------------
// Depthwise conv1d (k=4, pad=2, sliced to L) + bias + SiLU on ix half of XZ.
// -----------------------------------------------------------------------------
__global__ void conv_silu_kernel(const float* __restrict__ XZ,
                                 const float* __restrict__ cw,
                                 const float* __restrict__ cb,
                                 float* __restrict__ IX) {
  int idx = blockIdx.x * blockDim.x + threadIdx.x;
  if (idx >= MROWS * ED) return;
  int e = idx % ED;
  int r = idx / ED;
  int t = r % L_;
  int b = r / L_;
  float acc = cb[e];
  #pragma unroll
  for (int k = 0; k < 4; ++k) {
    int tt = t + k - 2;
    if (tt >= 0 && tt < L_)
      acc = fmaf(cw[e * 4 + k], XZ[((size_t)(b * L_ + tt)) * (2 * ED) + e], acc);
  }
  IX[idx] = acc / (1.f + __expf(-acc));   // silu
}

// -----------------------------------------------------------------------------
// delta[r,e] = softplus( sum_{j<12} DBC[r,j] * W_dt[j,e] + b_dt[e] )
// -----------------------------------------------------------------------------
__global__ void delta_kernel(const float* __restrict__ DBC,
                             const float* __restrict__ W_dt,
                             const float* __restrict__ b_dt,
                             float* __restrict__ DELTA) {
  int idx = blockIdx.x * blockDim.x + threadIdx.x;
  if (idx >= MROWS * ED) return;
  int e = idx % ED;
  int r = idx / ED;
  float acc = b_dt[e];
  #pragma unroll
  for (int j = 0; j < DTR; ++j)
    acc = fmaf(DBC[(size_t)r * DBCW + j], W_dt[j * ED + e], acc);
  DELTA[idx] = (acc > 20.f) ? acc : log1pf(__expf(acc));
}

// -----------------------------------------------------------------------------
// Selective scan, one thread per channel e (B*ED = 3072 threads total).
// h[64] and A[e,:] in VGPRs; B(t,:)/C(t,:) (shared across channels) staged
// 32 timesteps at a time in LDS (broadcast reads). Output fused:
//   Y2[r,e] = (y + D[e]*ix) * silu(z)
// -----------------------------------------------------------------------------
__global__ __launch_bounds__(256)
void scan_kernel(const float* __restrict__ XZ, const float* __restrict__ IX,
                 const float* __restrict__ DBC, const float* __restrict__ DELTA,
                 const float* __restrict__ A_log, const float* __restrict__ Dvec,
                 float* __restrict__ Y2) {
  __shared__ float Bsh[32][NST];
  __shared__ float Csh[32][NST];

  const int tid = threadIdx.x;
  const int e   = blockIdx.x * 256 + tid;
  const int b   = blockIdx.y;

  float Aa[NST], h[NST];
  #pragma unroll
  for (int n = 0; n < NST; ++n) {
    Aa[n] = -__expf(A_log[(size_t)e * NST + n]);
    h[n]  = 0.f;
  }
  const float De = Dvec[e];

  for (int t0 = 0; t0 < L_; t0 += 32) {
    #pragma unroll
    for (int i = 0; i < 16; ++i) {
      int idx = tid + i * 256;        // 0..4095
      int row = idx >> 7;
      int col = idx & 127;
      float v = DBC[(size_t)(b * L_ + t0 + row) * DBCW + DTR + col];
      if (col < NST) Bsh[row][col] = v;
      else           Csh[row][col - NST] = v;
    }
    __syncthreads();

    for (int tt = 0; tt < 32; ++tt) {
      const size_t r  = (size_t)(b * L_ + t0 + tt);
      const float dlt = DELTA[r * ED + e];
      const float ixv = IX[r * ED + e];
      const float zv  = XZ[r * (2 * ED) + ED + e];
      const float du  = dlt * ixv;
      float y = 0.f;
      #pragma unroll
      for (int n = 0; n < NST; ++n) {
        float dA = __expf(dlt * Aa[n]);
        h[n] = fmaf(dA, h[n], du * Bsh[tt][n]);
        y = fmaf(h[n], Csh[tt][n], y);
      }
      Y2[r * ED + e] = (y + De * ixv) * (zv / (1.f + __expf(-zv)));
    }
    __syncthreads();
  }
}

// -----------------------------------------------------------------------------
// Launch: GEMM1 -> conv+silu -> GEMM2 -> delta -> scan -> GEMM3
// Workspace usage (~64.1 MB)
// -----------------------------------------------------------------------------
extern "C" void kernel_launch(void* const* d_in, const int* in_sizes, int n_in,
                              void* d_out, int out_size, void* d_ws, size_t ws_size,
                              hipStream_t stream) {
  const float* x      = (const float*)d_in[0];
  const float* W_in   = (const float*)d_in[1];
  const float* conv_w = (const float*)d_in[2];
  const float* conv_b = (const float*)d_in[3];
  const float* W_x    = (const float*)d_in[4];
  const float* W_dt   = (const float*)d_in[5];
  const float* b_dt   = (const float*)d_in[6];
  const float* A_log  = (const float*)d_in[7];
  const float* Dv     = (const float*)d_in[8];
  const float* W_out  = (const float*)d_in[9];
  float* out = (float*)d_out;

  char* ws = (char*)d_ws;
  float* XZ    = (float*)ws;  ws += (size_t)MROWS * 2 * ED * sizeof(float);
  float* IX    = (float*)ws;  ws += (size_t)MROWS * ED * sizeof(float);
  float* DBCp  = (float*)ws;  ws += (size_t)MROWS * DBCW * sizeof(float);
  float* DELTA = (float*)ws;  ws += (size_t)MROWS * ED * sizeof(float);
  float* Y2    = (float*)ws;  ws += (size_t)MROWS * ED * sizeof(float);

  const int elems = MROWS * ED;

  // 1) XZ = x @ W_in   (2048 x 3072, K=768)
  gemm_f32_wmma<<<dim3((2 * ED + BN - 1) / BN, MROWS / BM), GT, 0, stream>>>(
      x, W_in, XZ, MROWS, 2 * ED, DM);

  // 2) IX = silu(depthwise_conv(ix) + b)
  conv_silu_kernel<<<(elems + 255) / 256, 256, 0, stream>>>(XZ, conv_w, conv_b, IX);

  // 3) DBC = IX @ W_x  (2048 x 140, K=1536)
  gemm_f32_wmma<<<dim3((DBCW + BN - 1) / BN, MROWS / BM), GT, 0, stream>>>(
      IX, W_x, DBCp, MROWS, DBCW, ED);

  // 4) DELTA = softplus(dt @ W_dt + b_dt)
  delta_kernel<<<(elems + 255) / 256, 256, 0, stream>>>(DBCp, W_dt, b_dt, DELTA);

  // 5) Y2 = (scan(...) + D*ix) * silu(z)
  scan_kernel<<<dim3(ED / 256, B_), 256, 0, stream>>>(
      XZ, IX, DBCp, DELTA, A_log, Dv, Y2);

  // 6) out = Y2 @ W_out (2048 x 768, K=1536)
  gemm_f32_wmma<<<dim3(DM / BN, MROWS / BM), GT, 0, stream>>>(
      Y2, W_out, out, MROWS, DM, ED);
}